// CRInstanceLoss_60189671686818
// MI455X (gfx1250) — compile-verified
//
#include <hip/hip_runtime.h>
#include <math.h>

typedef float v2f __attribute__((ext_vector_type(2)));
typedef float v8f __attribute__((ext_vector_type(8)));

constexpr int BN   = 512;   // batch
constexpr int NC   = 128;   // embedding dim == nclass
constexpr int KTOP = 8;
constexpr int BOUND = 4;    // int(P * B) = int(512/128)

// ---------------- row squared norms ----------------
__global__ void k_sq(const float* __restrict__ x, float* __restrict__ sq) {
  int i = blockIdx.x * blockDim.x + threadIdx.x;
  if (i < BN) {
    const float* row = x + i * NC;
    float s = 0.f;
    #pragma unroll 4
    for (int k = 0; k < NC; ++k) s += row[k] * row[k];
    sq[i] = s;
  }
}

// ---------------- zero word buffer ----------------
__global__ void k_zero(unsigned int* __restrict__ p, int nwords) {
  int i = blockIdx.x * blockDim.x + threadIdx.x;
  if (i < nwords) p[i] = 0u;
}

// ---------------- class counts -> anchor mask; zero accumulators ----------------
__global__ void k_anchors(const int* __restrict__ tgt, unsigned char* __restrict__ anch,
                          float* __restrict__ acc) {
  __shared__ int cnt[NC];
  int t = threadIdx.x;            // 512 threads, 1 block
  if (t < NC) cnt[t] = 0;
  __syncthreads();
  atomicAdd(&cnt[tgt[t]], 1);
  __syncthreads();
  anch[t] = (cnt[tgt[t]] < BOUND) ? 1 : 0;
  if (t < 2) acc[t] = 0.f;
}

// ---------------- d = pairwise Euclidean distances via WMMA f32 16x16x4 ----------------
// One wave (32 lanes) per 16x16 tile of dot = x @ x^T; K looped in steps of 4.
__global__ void k_dist_wmma(const float* __restrict__ x, const float* __restrict__ sq,
                            float* __restrict__ d) {
  const int lane = threadIdx.x;          // 0..31
  const int half = lane >> 4;            // 0 or 1
  const int r    = lane & 15;
  const int tm = blockIdx.y, tn = blockIdx.x;

  // A lane (half,r): row M=r of A-tile, K = k0 + 2*half (+1). B symmetric (N=r).
  const float* arow = x + (tm * 16 + r) * NC + 2 * half;
  const float* brow = x + (tn * 16 + r) * NC + 2 * half;

  v8f c = {};
  #pragma unroll
  for (int k0 = 0; k0 < NC; k0 += 4) {
    v2f a = *(const v2f*)(arow + k0);
    v2f b = *(const v2f*)(brow + k0);
    // (neg_a, A, neg_b, B, c_mod, C, reuse_a, reuse_b)
    c = __builtin_amdgcn_wmma_f32_16x16x4_f32(false, a, false, b, (short)0, c,
                                              false, false);
  }

  // C/D layout: VGPR v, lanes 0-15 -> M=v, N=lane; lanes 16-31 -> M=v+8, N=lane-16
  const int n = tn * 16 + r;
  const float sn = sq[n];
  #pragma unroll
  for (int v = 0; v < 8; ++v) {
    const int m = tm * 16 + v + 8 * half;
    float raw = sn - 2.0f * c[v] + sq[m];
    raw = fmaxf(raw, 0.0f);
    d[m * BN + n] = (raw == 0.0f) ? 0.0f : sqrtf(raw);   // zmask sqrt-safety trick
  }
}

// ---------------- per-column top-8 hard positive / hard negative masks ----------------
// Mirrors lax.top_k semantics: strict '>' scan ascending => smallest-index tie-break;
// -inf "selections" are never marked (they'd be removed by &sim / &~sim anyway).
__global__ void k_topk(const float* __restrict__ d, const int* __restrict__ tgt,
                       unsigned char* __restrict__ hp, unsigned char* __restrict__ hn) {
  int j = blockIdx.x * blockDim.x + threadIdx.x;
  if (j >= BN) return;
  const int tj = tgt[j];
  unsigned used[BN / 32];

  // hard positives: largest same-class distances in column j
  for (int w = 0; w < BN / 32; ++w) used[w] = 0u;
  for (int p = 0; p < KTOP; ++p) {
    float best = -INFINITY; int bi = -1;
    for (int i = 0; i < BN; ++i) {
      if (used[i >> 5] & (1u << (i & 31))) continue;
      float v = (tgt[i] == tj) ? d[i * BN + j] : -INFINITY;
      if (v > best) { best = v; bi = i; }
    }
    if (bi < 0) break;                       // only -inf left
    used[bi >> 5] |= 1u << (bi & 31);
    hp[bi * BN + j] = 1;                     // bi is same-class by construction
  }

  // hard negatives: smallest different-class distances in column j
  for (int w = 0; w < BN / 32; ++w) used[w] = 0u;
  for (int p = 0; p < KTOP; ++p) {
    float best = -INFINITY; int bi = -1;
    for (int i = 0; i < BN; ++i) {
      if (used[i >> 5] & (1u << (i & 31))) continue;
      float v = (tgt[i] == tj) ? -INFINITY : -d[i * BN + j];
      if (v > best) { best = v; bi = i; }
    }
    if (bi < 0) break;
    used[bi >> 5] |= 1u << (bi & 31);
    hn[bi * BN + j] = 1;
  }
}

// ---------------- sparse triplet reduction: one block per anchor row ----------------
__global__ void k_loss(const float* __restrict__ d, const unsigned char* __restrict__ hp,
                       const unsigned char* __restrict__ hn,
                       const unsigned char* __restrict__ anch, float* __restrict__ acc) {
  const int i = blockIdx.x;
  if (!anch[i]) return;                      // uniform across block

  __shared__ unsigned short apl[BN];
  __shared__ unsigned short anl[BN];
  __shared__ int nj, nk;
  __shared__ float ssum;
  __shared__ unsigned int scnt;
  if (threadIdx.x == 0) { nj = 0; nk = 0; ssum = 0.f; scnt = 0u; }
  __syncthreads();

  for (int j = threadIdx.x; j < BN; j += blockDim.x) {
    if (hp[i * BN + j] && j != i) { int p = atomicAdd(&nj, 1); apl[p] = (unsigned short)j; }
    if (hn[i * BN + j])           { int p = atomicAdd(&nk, 1); anl[p] = (unsigned short)j; }
  }
  __syncthreads();

  const int tot = nj * nk;
  const float* drow = d + i * BN;
  float lsum = 0.f; unsigned int lcnt = 0u;
  for (int p = threadIdx.x; p < tot; p += blockDim.x) {
    const int j = apl[p / nk];
    const int k = anl[p % nk];
    float t = drow[j] - drow[k] + 1.0f;      // MARGIN
    t = fmaxf(t, 0.0f);
    lsum += t;
    if (t > 1e-7f) ++lcnt;
  }
  atomicAdd(&ssum, lsum);
  atomicAdd(&scnt, lcnt);
  __syncthreads();
  if (threadIdx.x == 0) {
    atomicAdd(&acc[0], ssum);
    atomicAdd(&acc[1], (float)scnt);
  }
}

__global__ void k_final(const float* __restrict__ acc, float* __restrict__ out) {
  out[0] = acc[0] / (acc[1] + 1e-7f);
}

// ---------------- host launcher ----------------
extern "C" void kernel_launch(void* const* d_in, const int* in_sizes, int n_in,
                              void* d_out, int out_size, void* d_ws, size_t ws_size,
                              hipStream_t stream) {
  const float* x   = (const float*)d_in[0];   // (512, 128) f32
  const int*   tgt = (const int*)d_in[1];     // (512,) i32
  float* out = (float*)d_out;

  // workspace layout (bytes)
  char* ws = (char*)d_ws;
  float*         sq   = (float*)(ws + 0);                     //   512 f32
  float*         d    = (float*)(ws + 2048);                  // 512*512 f32 (1 MB)
  unsigned char* hp   = (unsigned char*)(ws + 1050624);       // 512*512 u8
  unsigned char* hn   = (unsigned char*)(ws + 1312768);       // 512*512 u8
  unsigned char* anch = (unsigned char*)(ws + 1574912);       //   512 u8
  float*         acc  = (float*)(ws + 1575424);               // {sum, count}

  k_sq<<<dim3((BN + 255) / 256), dim3(256), 0, stream>>>(x, sq);
  {
    const int nwords = (2 * BN * BN) / 4;                     // hp+hn contiguous
    k_zero<<<dim3((nwords + 255) / 256), dim3(256), 0, stream>>>((unsigned int*)hp, nwords);
  }
  k_anchors<<<dim3(1), dim3(BN), 0, stream>>>(tgt, anch, acc);
  k_dist_wmma<<<dim3(BN / 16, BN / 16), dim3(32), 0, stream>>>(x, sq, d);
  k_topk<<<dim3((BN + 255) / 256), dim3(256), 0, stream>>>(d, tgt, hp, hn);
  k_loss<<<dim3(BN), dim3(256), 0, stream>>>(d, hp, hn, anch, acc);
  k_final<<<dim3(1), dim3(1), 0, stream>>>(acc, out);
  (void)in_sizes; (void)n_in; (void)out_size; (void)ws_size;
}